// MockStreamGenerator_51161650430413
// MI455X (gfx1250) — compile-verified
//
#include <hip/hip_runtime.h>
#include <hip/hip_bf16.h>
#include <stdint.h>

// Softened-Kepler leapfrog integrator, GM = 1, A_SCALE = 1.
// CDNA5 path: per-block TDM (tensor_load_to_lds) stages the two 3072B input
// row tiles into LDS; compute keeps the 6-float state in VGPRs for all steps.

#define BLOCK 256
#define ROWS_PER_BLOCK (BLOCK / 2)          // 128 particles per buffer per block
#define TILE_ELEMS (ROWS_PER_BLOCK * 6)     // 768 floats per buffer tile
#define TILE_BYTES (TILE_ELEMS * 4)         // 3072 bytes

typedef unsigned int u32x4 __attribute__((ext_vector_type(4)));
typedef int          i32x4 __attribute__((ext_vector_type(4)));
typedef int          i32x8 __attribute__((ext_vector_type(8)));

typedef __attribute__((address_space(3))) float lds_float_t;

// Issue one TDM load of a contiguous run of `valid_elems` f32 (zero-filled up
// to TILE_ELEMS via tensor OOB semantics) from `gptr` into LDS byte offset
// `lds_off`. Wave-level op; EXEC ignored; tracked by TENSORcnt.
__device__ __forceinline__ void tdm_load_tile(uint32_t lds_off, const float* gptr,
                                              uint32_t valid_elems) {
    const uint64_t ga = (uint64_t)(uintptr_t)gptr;

    // ---- D# group 0 (128b): count=1 | lds_addr | global_addr[56:0] | type=2
    u32x4 g0;
    g0.x = 1u;                                        // count=1, is_restore=0, no gather
    g0.y = lds_off;                                   // lds_addr (bytes)
    g0.z = (uint32_t)(ga & 0xFFFFFFFFu);              // global_addr[31:0]
    g0.w = (uint32_t)((ga >> 32) & 0x1FFFFFFu)        // global_addr[56:32]
         | (2u << 30);                                // type = 2 ("image")

    // ---- D# group 1 (256b): degenerate-2D descriptor, data_size=4B
    const uint32_t tile0 = (uint32_t)TILE_ELEMS;      // tile_dim0 (<= 16 bits)
    uint32_t d0 = 2u << 16;                           // data_size=2 (4B); mask/pad/iter=0
    uint32_t d1 = (valid_elems & 0xFFFFu) << 16;      // tensor_dim0[15:0]  -> bits 63:48
    uint32_t d2 = ((valid_elems >> 16) & 0xFFFFu)     // tensor_dim0[31:16] -> bits 79:64
                | (1u << 16);                         // tensor_dim1 = 1    -> bits 95:80
    uint32_t d3 = (tile0 << 16);                      // tile_dim0          -> bits 127:112
    uint32_t d4 = 1u;                                 // tile_dim1 = 1      -> bits 143:128
    uint32_t d5 = tile0;                              // tensor_dim0_stride[31:0]
    uint32_t d6 = 0u;                                 // stride[47:32] | tensor_dim1_stride lo
    uint32_t d7 = 0u;
    i32x8 g1 = { (int)d0, (int)d1, (int)d2, (int)d3,
                 (int)d4, (int)d5, (int)d6, (int)d7 };

    i32x4 g2 = { 0, 0, 0, 0 };                        // unused (<=2D tensor)
    i32x4 g3 = { 0, 0, 0, 0 };

#if defined(__clang_major__) && __clang_major__ >= 23
    i32x8 gx = { 0, 0, 0, 0, 0, 0, 0, 0 };
    __builtin_amdgcn_tensor_load_to_lds(g0, g1, g2, g3, gx, 0);
#else
    __builtin_amdgcn_tensor_load_to_lds(g0, g1, g2, g3, 0);
#endif
}

__device__ __forceinline__ void accel_soft(float qx, float qy, float qz,
                                           float& ax, float& ay, float& az) {
    // r = sqrt(q.q); a = -q / (r*(r+1)^2 + 1e-12)
    float r2 = fmaf(qx, qx, fmaf(qy, qy, qz * qz));
    float r  = __builtin_amdgcn_sqrtf(r2);            // raw v_sqrt_f32
    float s  = r + 1.0f;                              // A_SCALE = 1
    float den = fmaf(r, s * s, 1e-12f);
    float inv = __builtin_amdgcn_rcpf(den);           // raw v_rcp_f32
    ax = -qx * inv;                                   // GM = 1
    ay = -qy * inv;
    az = -qz * inv;
}

__global__ void __launch_bounds__(BLOCK)
leapfrog_kernel(const float* __restrict__ ts,
                const float* __restrict__ w0_lead,
                const float* __restrict__ w0_trail,
                const int*   __restrict__ n_steps_p,
                float*       __restrict__ out,
                int n)  // n = N particles; output rows = 2n
{
    // LDS tile: [0, TILE_BYTES) = lead rows, [TILE_BYTES, 2*TILE_BYTES) = trail rows
    __shared__ __align__(16) float smem[2 * TILE_ELEMS];

    const int tid  = blockIdx.x * BLOCK + threadIdx.x;
    const int base = blockIdx.x * ROWS_PER_BLOCK;     // first particle of this block

    // ---- Stage both input tiles into LDS with the Tensor Data Mover.
    // One wave issues both DMAs (TDM ignores EXEC; guard avoids duplicates).
    if (threadIdx.x == 0) {
        const uint32_t lds_base =
            (uint32_t)(uintptr_t)(lds_float_t*)&smem[0];
        const int rows_avail = n - base;
        const uint32_t valid =
            (uint32_t)((rows_avail >= ROWS_PER_BLOCK ? ROWS_PER_BLOCK : rows_avail) * 6);
        tdm_load_tile(lds_base,              w0_lead  + 6 * (size_t)base, valid);
        tdm_load_tile(lds_base + TILE_BYTES, w0_trail + 6 * (size_t)base, valid);
        __builtin_amdgcn_s_wait_tensorcnt(0);         // s_wait_tensorcnt 0
    }
    __syncthreads();                                  // publish LDS tiles to all waves

    if (tid < 2 * n) {
        const int i       = tid >> 1;                 // particle index
        const int li      = threadIdx.x >> 1;         // particle index within block
        const int is_lead = tid & 1;                  // 1 = lead (odd out rows)

        // Uniform-address reads -> scalar loads.
        const int   nsteps = n_steps_p[0];
        const float t_f    = ts[n - 1] + 0.001f;

        const float dt  = (t_f - ts[i]) / (float)nsteps;
        const float hdt = 0.5f * dt;

        // Row from LDS: 24B, 8B-aligned -> three ds_load_b64.
        const float* lrow = &smem[(is_lead ? 0 : TILE_ELEMS) + 6 * li];
        const float2* row = reinterpret_cast<const float2*>(lrow);
        float2 d01 = row[0];
        float2 d23 = row[1];
        float2 d45 = row[2];

        float qx = d01.x, qy = d01.y, qz = d23.x;
        float px = d23.y, py = d45.x, pz = d45.y;

        float ax, ay, az;

#pragma unroll 4
        for (int s = 0; s < nsteps; ++s) {
            // kick (half)
            accel_soft(qx, qy, qz, ax, ay, az);
            float phx = fmaf(hdt, ax, px);
            float phy = fmaf(hdt, ay, py);
            float phz = fmaf(hdt, az, pz);
            // drift
            qx = fmaf(dt, phx, qx);
            qy = fmaf(dt, phy, qy);
            qz = fmaf(dt, phz, qz);
            // kick (half)
            accel_soft(qx, qy, qz, ax, ay, az);
            px = fmaf(hdt, ax, phx);
            py = fmaf(hdt, ay, phy);
            pz = fmaf(hdt, az, phz);
        }

        // Coalesced 24B store per thread: three global_store_b64.
        float2* __restrict__ orow = reinterpret_cast<float2*>(out + 6 * (size_t)tid);
        orow[0] = make_float2(qx, qy);
        orow[1] = make_float2(qz, px);
        orow[2] = make_float2(py, pz);
    }
}

extern "C" void kernel_launch(void* const* d_in, const int* in_sizes, int n_in,
                              void* d_out, int out_size, void* d_ws, size_t ws_size,
                              hipStream_t stream) {
    const float* ts       = (const float*)d_in[0];
    const float* w0_lead  = (const float*)d_in[1];
    const float* w0_trail = (const float*)d_in[2];
    const int*   n_steps  = (const int*)d_in[3];
    float*       out      = (float*)d_out;

    const int n = in_sizes[0];              // N particles
    const int total = 2 * n;                // output rows / threads
    const int grid = (total + BLOCK - 1) / BLOCK;

    leapfrog_kernel<<<grid, BLOCK, 0, stream>>>(ts, w0_lead, w0_trail, n_steps, out, n);
}